// classifier_80547816669785
// MI455X (gfx1250) — compile-verified
//
#include <hip/hip_runtime.h>
#include <stdint.h>

#define NN 50000
#define NE 800000
#define NG 64
#define DH 128
#define BN_EPS 1e-5f
#define KEEP_P 0.8f

typedef __attribute__((ext_vector_type(2))) float v2f;
typedef __attribute__((ext_vector_type(8))) float v8f;

#if defined(__HIP_DEVICE_COMPILE__) && __has_builtin(__builtin_amdgcn_global_load_async_to_lds_b128) && __has_builtin(__builtin_amdgcn_s_wait_asynccnt)
#define HAVE_ASYNC_LDS 1
typedef int v4i_g __attribute__((vector_size(16)));          // matches builtin's param type
typedef __attribute__((address_space(1))) v4i_g as1_v4i;     // global (__device__)
typedef __attribute__((address_space(3))) v4i_g as3_v4i;     // LDS (__shared__)
#else
#define HAVE_ASYNC_LDS 0
#endif

// ---------------- threefry2x32 (exact JAX semantics) ----------------
__host__ __device__ __forceinline__ uint32_t rotl32(uint32_t x, uint32_t r) {
    return (x << r) | (x >> (32u - r));
}

__host__ __device__ __forceinline__ void threefry2x32(uint32_t k0, uint32_t k1,
                                                      uint32_t& x0, uint32_t& x1) {
    const uint32_t ks2 = k0 ^ k1 ^ 0x1BD11BDAu;
    const uint32_t ks[3] = {k0, k1, ks2};
    x0 += k0; x1 += k1;
    const uint32_t rot[5][4] = {{13,15,26,6},{17,29,16,24},{13,15,26,6},{17,29,16,24},{13,15,26,6}};
#pragma unroll
    for (int g = 0; g < 5; ++g) {
#pragma unroll
        for (int r = 0; r < 4; ++r) { x0 += x1; x1 = rotl32(x1, rot[g][r]); x1 ^= x0; }
        x0 += ks[(g + 1) % 3];
        x1 += ks[(g + 2) % 3] + (uint32_t)(g + 1);
    }
}

// ---------------- elementwise / utility kernels ----------------
__global__ void k_zero(float* __restrict__ p, int n) {
    int i = blockIdx.x * blockDim.x + threadIdx.x;
    if (i < n) p[i] = 0.f;
}

// z = (1+eps) * h   (vectorized)
__global__ void k_init_z(const float4* __restrict__ h, const float* __restrict__ eps,
                         float4* __restrict__ z, int n4) {
    int i = blockIdx.x * blockDim.x + threadIdx.x;
    if (i < n4) {
        float s = 1.f + eps[0];
        float4 v = h[i];
        v.x *= s; v.y *= s; v.z *= s; v.w *= s;
        z[i] = v;
    }
}

// z[dst] += h[src]  over edges, float4 chunks, native f32 atomics
__global__ void k_scatter(const float* __restrict__ h, const int* __restrict__ src,
                          const int* __restrict__ dst, float* __restrict__ z,
                          int D, int chunks) {
    int tid = blockIdx.x * blockDim.x + threadIdx.x;
    int total = NE * chunks;
    if (tid >= total) return;
    int e = tid / chunks;
    int c = (tid - e * chunks) * 4;
    int s = src[e], d = dst[e];
    const float4 v = *(const float4*)(h + (size_t)s * D + c);
    float* zp = z + (size_t)d * D + c;
    unsafeAtomicAdd(zp + 0, v.x);
    unsafeAtomicAdd(zp + 1, v.y);
    unsafeAtomicAdd(zp + 2, v.z);
    unsafeAtomicAdd(zp + 3, v.w);
}

// ---------------- f32 WMMA GEMM: C[MxN] = A[MxK] @ W[KxN] ----------------
// One 16x64 output tile per wave32 (4 accumulators); W staged in LDS via
// async global->LDS loads; requires 16|M, 64|N, 4|K.
#define WMMA4(a, b, acc) \
    (acc) = __builtin_amdgcn_wmma_f32_16x16x4_f32(false, (a), false, (b), (short)0, (acc), false, false)

__global__ void k_gemm_wmma(const float* __restrict__ A, const float* __restrict__ W,
                            float* __restrict__ C, int M, int K, int N) {
    extern __shared__ float ws[];                 // K*N floats
    const int t = threadIdx.x;
    const int n4 = (K * N) >> 2;
#if HAVE_ASYNC_LDS
    for (int i = t; i < n4; i += blockDim.x)
        __builtin_amdgcn_global_load_async_to_lds_b128(
            (as1_v4i*)(((const float4*)W) + i),
            (as3_v4i*)(((float4*)ws) + i), 0, 0);
    __builtin_amdgcn_s_wait_asynccnt(0);
#else
    for (int i = t; i < n4; i += blockDim.x)
        ((float4*)ws)[i] = ((const float4*)W)[i];
#endif
    __syncthreads();

    const int wave = t >> 5;
    const int lane = t & 31;
    const int mTiles = M >> 4;
    const int nGroups = N >> 6;                   // 64-wide column groups
    const int tile = blockIdx.x * (blockDim.x >> 5) + wave;
    if (tile >= mTiles * nGroups) return;         // wave-uniform: EXEC stays all-1
    const int tm = tile % mTiles, tn = tile / mTiles;

    const int hi = lane >> 4;                     // half-wave select
    const int lo = lane & 15;
    // A fragment base: row M = tm*16+lo, columns k0+2*hi, k0+2*hi+1
    const float* Ap = A + (size_t)(tm * 16 + lo) * K + 2 * hi;
    // B fragment base: column tn*64 + sub*16 + lo, rows k0+2*hi (+1)
    const float* Bl = ws + tn * 64 + lo + (size_t)(2 * hi) * N;

    v8f acc0 = {0.f,0.f,0.f,0.f,0.f,0.f,0.f,0.f};
    v8f acc1 = {0.f,0.f,0.f,0.f,0.f,0.f,0.f,0.f};
    v8f acc2 = {0.f,0.f,0.f,0.f,0.f,0.f,0.f,0.f};
    v8f acc3 = {0.f,0.f,0.f,0.f,0.f,0.f,0.f,0.f};
    for (int k0 = 0; k0 < K; k0 += 4) {
        v2f a;
        a.x = Ap[k0 + 0];
        a.y = Ap[k0 + 1];
        const float* Bk = Bl + (size_t)k0 * N;
        v2f b0, b1, b2, b3;
        b0.x = Bk[0];       b0.y = Bk[N + 0];
        b1.x = Bk[16];      b1.y = Bk[N + 16];
        b2.x = Bk[32];      b2.y = Bk[N + 32];
        b3.x = Bk[48];      b3.y = Bk[N + 48];
        WMMA4(a, b0, acc0);
        WMMA4(a, b1, acc1);
        WMMA4(a, b2, acc2);
        WMMA4(a, b3, acc3);
    }
    // C/D layout: VGPR r -> M = r + 8*hi, N = lo (+ sub*16)
    float* Ct = C + (size_t)(tm * 16) * N + tn * 64;
#pragma unroll
    for (int r = 0; r < 8; ++r) {
        float* row = Ct + (size_t)(r + 8 * hi) * N + lo;
        row[0]  = acc0[r];
        row[16] = acc1[r];
        row[32] = acc2[r];
        row[48] = acc3[r];
    }
}

// small GEMM for the last layer (N=2)
__global__ void k_gemm2(const float* __restrict__ A, const float* __restrict__ W,
                        float* __restrict__ C, int M, int K) {
    int r = blockIdx.x * blockDim.x + threadIdx.x;
    if (r >= M) return;
    const float* a = A + (size_t)r * K;
    float s0 = 0.f, s1 = 0.f;
    for (int k = 0; k < K; ++k) {
        float av = a[k];
        s0 += av * W[k * 2 + 0];
        s1 += av * W[k * 2 + 1];
    }
    C[r * 2 + 0] = s0;
    C[r * 2 + 1] = s1;
}

// column sums + sums of squares (for biased batch stats)
__global__ void k_colstats(const float* __restrict__ X, int M, int N,
                           float* __restrict__ stats) {
    int t = blockIdx.x * blockDim.x + threadIdx.x;       // 65536 threads, N | 65536
    int c = t % N;
    int nper = (gridDim.x * blockDim.x) / N;
    float s = 0.f, sq = 0.f;
    for (int r = t / N; r < M; r += nper) {
        float v = X[(size_t)r * N + c];
        s += v; sq += v * v;
    }
    unsafeAtomicAdd(&stats[c], s);
    unsafeAtomicAdd(&stats[N + c], sq);
}

// in-place BN (training stats) + ReLU
__global__ void k_bn_relu(float* __restrict__ X, const float* __restrict__ stats,
                          const float* __restrict__ g, const float* __restrict__ b,
                          int M, int N) {
    int i = blockIdx.x * blockDim.x + threadIdx.x;
    if (i >= M * N) return;
    int c = i % N;
    float inv = 1.f / (float)M;
    float mu = stats[c] * inv;
    float var = stats[N + c] * inv - mu * mu;
    float v = g[c] * (X[i] - mu) * rsqrtf(var + BN_EPS) + b[c];
    X[i] = fmaxf(v, 0.f);
}

// optional BN -> JAX dropout(p=0.2) -> ReLU, written to `out`
__global__ void k_bn_drop_relu(const float* __restrict__ X, const float* __restrict__ stats,
                               const float* __restrict__ g, const float* __restrict__ b,
                               float* __restrict__ out, int M, int N, int use_bn,
                               uint32_t k0, uint32_t k1) {
    int i = blockIdx.x * blockDim.x + threadIdx.x;
    int total = M * N;
    if (i >= total) return;
    float v = X[i];
    if (use_bn) {
        int c = i % N;
        float inv = 1.f / (float)M;
        float mu = stats[c] * inv;
        float var = stats[N + c] * inv - mu * mu;
        v = g[c] * (v - mu) * rsqrtf(var + BN_EPS) + b[c];
    }
    // threefry_2x32(key, iota(total)): pairs (p, p+half), lane select by i<half
    int half = total >> 1;
    uint32_t p = (uint32_t)(i < half ? i : i - half);
    uint32_t x0 = p, x1 = p + (uint32_t)half;
    threefry2x32(k0, k1, x0, x1);
    uint32_t bits = (i < half) ? x0 : x1;
    float u = __uint_as_float((bits >> 9) | 0x3f800000u) - 1.0f;
    v = (u < KEEP_P) ? v * (1.0f / KEEP_P) : 0.0f;
    out[i] = fmaxf(v, 0.f);
}

// ---------------- global mean pool ----------------
__global__ void k_pool_acc(const float* __restrict__ h, const int* __restrict__ batch,
                           float* __restrict__ sums, float* __restrict__ cnt) {
    int i = blockIdx.x * blockDim.x + threadIdx.x;
    if (i >= NN) return;
    int gidx = batch[i];
    unsafeAtomicAdd(&sums[gidx * 2 + 0], h[i * 2 + 0]);
    unsafeAtomicAdd(&sums[gidx * 2 + 1], h[i * 2 + 1]);
    unsafeAtomicAdd(&cnt[gidx], 1.f);
}

__global__ void k_pool_final(const float* __restrict__ sums, const float* __restrict__ cnt,
                             float* __restrict__ out) {
    int i = blockIdx.x * blockDim.x + threadIdx.x;
    if (i < NG * 2) out[i] = sums[i] / fmaxf(cnt[i >> 1], 1.f);
}

// ---------------- host driver ----------------
extern "C" void kernel_launch(void* const* d_in, const int* in_sizes, int n_in,
                              void* d_out, int out_size, void* d_ws, size_t ws_size,
                              hipStream_t stream) {
    (void)in_sizes; (void)n_in; (void)out_size; (void)ws_size;
    const float* x   = (const float*)d_in[0];
    const int* ei    = (const int*)d_in[1];
    const int* srcE  = ei;
    const int* dstE  = ei + NE;
    const int* batch = (const int*)d_in[2];

    const float *eps[7], *W1[7], *g1[7], *b1[7], *W2[7], *g2[7], *b2[7];
    int idx = 3;
    for (int li = 0; li < 7; ++li) {
        eps[li] = (const float*)d_in[idx++];
        W1 [li] = (const float*)d_in[idx++];
        g1 [li] = (const float*)d_in[idx++];
        b1 [li] = (const float*)d_in[idx++];
        W2 [li] = (const float*)d_in[idx++];
        if (li < 6) { g2[li] = (const float*)d_in[idx++]; b2[li] = (const float*)d_in[idx++]; }
        else        { g2[li] = nullptr; b2[li] = nullptr; }
    }

    // workspace layout (floats)
    float* ws   = (float*)d_ws;
    float* z    = ws;                                   // 50000*128
    float* y1   = z    + (size_t)NN * 128;              // 50000*128
    float* y2   = y1   + (size_t)NN * 128;              // 50000*64
    float* hbuf = y2   + (size_t)NN * 64;               // 50000*64
    float* st1  = hbuf + (size_t)NN * 64;               // 256
    float* st2  = st1 + 256;                            // 256
    float* psum = st2 + 256;                            // 128
    float* pcnt = psum + 128;                           // 64

    // dropout keys: fold_in(key(42), li) on host (deterministic, capture-safe)
    uint32_t dk0[7], dk1[7];
    for (int li = 0; li < 7; ++li) {
        uint32_t a = 0u, b = (uint32_t)li;
        threefry2x32(0u, 42u, a, b);
        dk0[li] = a; dk1[li] = b;
    }

    const int din[7]  = {128, 64, 64, 64, 64, 64, 64};
    const int dout[7] = {64, 64, 64, 64, 64, 64, 2};

    const float* hcur = x;
    for (int li = 0; li < 7; ++li) {
        const int D  = din[li];
        const int DO = dout[li];
        const int nz = NN * D;

        k_init_z<<<(nz / 4 + 255) / 256, 256, 0, stream>>>(
            (const float4*)hcur, eps[li], (float4*)z, nz / 4);

        const int chunks = D / 4;
        const int etotal = NE * chunks;
        k_scatter<<<(etotal + 255) / 256, 256, 0, stream>>>(hcur, srcE, dstE, z, D, chunks);

        {   // y1 = z @ W1  (M=50000, K=D, N=128) — f32 WMMA, 16x64 per wave
            int waves = (NN / 16) * (DH / 64);
            k_gemm_wmma<<<(waves + 7) / 8, 256, (size_t)D * DH * 4, stream>>>(
                z, W1[li], y1, NN, D, DH);
        }
        k_zero<<<1, 256, 0, stream>>>(st1, 256);
        k_colstats<<<256, 256, 0, stream>>>(y1, NN, DH, st1);
        k_bn_relu<<<(NN * DH + 255) / 256, 256, 0, stream>>>(y1, st1, g1[li], b1[li], NN, DH);

        if (li < 6) {
            int waves = (NN / 16) * (DO / 64);
            k_gemm_wmma<<<(waves + 7) / 8, 256, (size_t)DH * DO * 4, stream>>>(
                y1, W2[li], y2, NN, DH, DO);
            k_zero<<<1, 256, 0, stream>>>(st2, 256);
            k_colstats<<<256, 256, 0, stream>>>(y2, NN, DO, st2);
            k_bn_drop_relu<<<(NN * DO + 255) / 256, 256, 0, stream>>>(
                y2, st2, g2[li], b2[li], hbuf, NN, DO, 1, dk0[li], dk1[li]);
        } else {
            k_gemm2<<<(NN + 255) / 256, 256, 0, stream>>>(y1, W2[li], y2, NN, DH);
            k_bn_drop_relu<<<(NN * DO + 255) / 256, 256, 0, stream>>>(
                y2, nullptr, nullptr, nullptr, hbuf, NN, DO, 0, dk0[li], dk1[li]);
        }
        hcur = hbuf;
    }

    // global mean pool -> d_out (64 x 2, f32)
    k_zero<<<1, 256, 0, stream>>>(psum, 128 + 64);
    k_pool_acc<<<(NN + 255) / 256, 256, 0, stream>>>(hcur, batch, psum, pcnt);
    k_pool_final<<<1, 128, 0, stream>>>(psum, pcnt, (float*)d_out);
}